// GCN_9594956939694
// MI455X (gfx1250) — compile-verified
//
#include <hip/hip_runtime.h>
#include <math.h>

typedef float v2f __attribute__((ext_vector_type(2)));
typedef float v8f __attribute__((ext_vector_type(8)));

#define WPB 8  // waves per block (256 threads, wave32)

// ---------------- degree / dinv ----------------
__global__ void k_init_deg(float* deg, int N) {
  int i = blockIdx.x * blockDim.x + threadIdx.x;
  if (i < N) deg[i] = 1.0f;  // self-loop
}

__global__ void k_deg(const int* __restrict__ dst, float* __restrict__ deg, long E) {
  long e = (long)blockIdx.x * blockDim.x + threadIdx.x;
  if (e < E) atomicAdd(&deg[dst[e]], 1.0f);
}

__global__ void k_dinv(float* deg, int N) {
  int i = blockIdx.x * blockDim.x + threadIdx.x;
  if (i < N) {
    float d = deg[i];
    deg[i] = (d > 0.0f) ? rsqrtf(d) : 0.0f;
  }
}

// ---------------- WMMA GEMM: G = (X @ W) * dinv[row] ----------------
// Block = 8 row-tiles x 1 col-tile. The 16-column W slab (Cin x 16) is staged
// once into LDS in a k-paired layout so each lane's B fragment is one
// bank-conflict-free ds_load_b64; the K-loop then issues 1 global b64 (A) +
// 1 LDS b64 (B) + 1 v_wmma_f32_16x16x4_f32 per step with immediate strides.
//
// Per-ISA VGPR layouts:
//  A (16x4 f32):  lane&15 = M row, kb = (lane>>4)*2 selects K pair {0,1}/{2,3}
//  B (4x16 f32):  lane&15 = N col, same kb pairing
//  C/D (16x16):   VGPR v -> M = v + 8*(lane>=16), N = lane&15
template <int Cin, int Cout>
__global__ __launch_bounds__(256) void k_gemm_wmma(
    const float* __restrict__ X, const float* __restrict__ W,
    const float* __restrict__ dinv, float* __restrict__ G, int nRowTiles) {
  __shared__ float Bs[Cin * 16];  // Bs[(k>>1)*32 + col*2 + (k&1)]

  const int tid  = threadIdx.x;
  const int lane = tid & 31;
  const int wave = tid >> 5;
  const int col0 = (int)blockIdx.y << 4;

  // cooperative stage of the B slab (W[:, col0:col0+16]) into LDS
  #pragma unroll
  for (int idx = tid; idx < Cin * 16; idx += 256) {
    int k = idx >> 4;
    int c = idx & 15;
    Bs[(k >> 1) * 32 + c * 2 + (k & 1)] = W[(long)k * Cout + (col0 + c)];
  }
  __syncthreads();

  int rowTile = (int)blockIdx.x * WPB + wave;
  const bool active = (rowTile < nRowTiles);
  if (!active) rowTile = nRowTiles - 1;  // clamp: keep wave computing, EXEC all-ones
  const int row0 = rowTile << 4;
  const int r  = lane & 15;
  const int kb = (lane >> 4) << 1;  // 0 or 2

  const float* __restrict__ Arow = X + (long)(row0 + r) * Cin + kb;
  const float* __restrict__ Bl   = Bs + (kb >> 1) * 32 + r * 2;

  v8f acc = {0.f, 0.f, 0.f, 0.f, 0.f, 0.f, 0.f, 0.f};
  #pragma unroll 8
  for (int k = 0; k < Cin; k += 4) {
    v2f a = *(const v2f*)(Arow + k);             // A[row][k+kb], A[row][k+kb+1]
    v2f b = *(const v2f*)(Bl + (k >> 1) * 32);   // B[k+kb][col], B[k+kb+1][col]
    acc = __builtin_amdgcn_wmma_f32_16x16x4_f32(
        /*neg_a=*/false, a, /*neg_b=*/false, b,
        /*c_mod=*/(short)0, acc, /*reuse_a=*/false, /*reuse_b=*/false);
  }

  if (!active) return;  // wave-uniform
  const int m0 = (lane >> 4) << 3;  // 0 or 8
  const int c  = col0 + r;
  #pragma unroll
  for (int v = 0; v < 8; ++v) {
    int rr = row0 + m0 + v;
    G[(long)rr * Cout + c] = acc[v] * dinv[rr];
  }
}

// ---------------- edge scatter: AGG[dst] += G[src] ----------------
// 32 lanes per edge, CH = Cout/32 channels per lane, f32 global atomics.
template <int CH>
__global__ void k_scatter(const int* __restrict__ src, const int* __restrict__ dst,
                          const float* __restrict__ G, float* __restrict__ AGG,
                          long E, int Cout) {
  long t = (long)blockIdx.x * blockDim.x + threadIdx.x;
  long e = t >> 5;
  if (e >= E) return;
  const int lane = (int)(t & 31);
  const int s = src[e];
  const int d = dst[e];
  const float* __restrict__ g = G + (long)s * Cout + lane * CH;
  float* a = AGG + (long)d * Cout + lane * CH;
  #pragma unroll
  for (int i = 0; i < CH; ++i) atomicAdd(a + i, g[i]);
}

// ---------------- epilogue: X' = relu(dinv[i]*(AGG + G) + b) ----------------
// (G term = analytic self-loop contribution; writes in-place over G)
__global__ void k_epilogue(const float* __restrict__ AGG, const float* __restrict__ G,
                           const float* __restrict__ dinv, const float* __restrict__ bias,
                           float* __restrict__ Xn, int Cout) {
  int i = blockIdx.x;
  int c = threadIdx.x;
  long idx = (long)i * Cout + c;
  float v = dinv[i] * (AGG[idx] + G[idx]) + bias[c];
  Xn[idx] = fmaxf(v, 0.0f);
}

// ---------------- final layer: bias + relu + log_softmax (Cout=64) ----------------
__global__ void k_epilogue_lsm(const float* __restrict__ AGG, const float* __restrict__ G,
                               const float* __restrict__ dinv, const float* __restrict__ bias,
                               float* __restrict__ out, int N) {
  const int lane = threadIdx.x & 31;
  const int wave = threadIdx.x >> 5;
  long i = (long)blockIdx.x * WPB + wave;
  if (i >= N) return;
  const float di = dinv[i];
  long base = i * 64;
  float v0 = fmaxf(di * (AGG[base + lane]      + G[base + lane])      + bias[lane],      0.0f);
  float v1 = fmaxf(di * (AGG[base + 32 + lane] + G[base + 32 + lane]) + bias[32 + lane], 0.0f);
  float m = fmaxf(v0, v1);
  #pragma unroll
  for (int o = 16; o > 0; o >>= 1) m = fmaxf(m, __shfl_xor(m, o, 32));
  float s = expf(v0 - m) + expf(v1 - m);
  #pragma unroll
  for (int o = 16; o > 0; o >>= 1) s += __shfl_xor(s, o, 32);
  float lse = m + logf(s);
  out[base + lane]      = v0 - lse;
  out[base + 32 + lane] = v1 - lse;
}

// ---------------- host launcher ----------------
extern "C" void kernel_launch(void* const* d_in, const int* in_sizes, int n_in,
                              void* d_out, int out_size, void* d_ws, size_t ws_size,
                              hipStream_t stream) {
  const float* x  = (const float*)d_in[0];
  const int*   ei = (const int*)d_in[1];
  const float* W1 = (const float*)d_in[2];
  const float* b1 = (const float*)d_in[3];
  const float* W2 = (const float*)d_in[4];
  const float* b2 = (const float*)d_in[5];
  const float* W3 = (const float*)d_in[6];
  const float* b3 = (const float*)d_in[7];

  const int C1o = in_sizes[3];         // 256
  const int C2o = in_sizes[5];         // 128
  const int C3o = in_sizes[7];         // 64
  const int C1i = in_sizes[2] / C1o;   // 128
  const int N   = in_sizes[0] / C1i;   // 100000
  const long E  = (long)in_sizes[1] / 2;

  const int* srcIdx = ei;
  const int* dstIdx = ei + E;

  // workspace layout (floats): dinv[N] | big0[N*C1o] | big1[N*C2o] | agg[N*C1o]
  float* dinv = (float*)d_ws;
  float* big0 = dinv + N;
  float* big1 = big0 + (long)N * C1o;
  float* agg  = big1 + (long)N * C2o;

  // degree + dinv
  k_init_deg<<<(N + 255) / 256, 256, 0, stream>>>(dinv, N);
  k_deg<<<(unsigned)((E + 255) / 256), 256, 0, stream>>>(dstIdx, dinv, E);
  k_dinv<<<(N + 255) / 256, 256, 0, stream>>>(dinv, N);

  const int nRT = N / 16;  // 6250 row tiles (N divisible by 16)
  const unsigned rowBlocks = (unsigned)((nRT + WPB - 1) / WPB);
  const long edgeThreads = E * 32;
  const unsigned edgeBlocks = (unsigned)((edgeThreads + 255) / 256);

  // ---- layer 1: x(128) @ W1 -> 256 ----
  {
    dim3 grid(rowBlocks, C1o / 16);
    k_gemm_wmma<128, 256><<<grid, 256, 0, stream>>>(x, W1, dinv, big0, nRT);
    hipMemsetAsync(agg, 0, (size_t)N * C1o * sizeof(float), stream);
    k_scatter<8><<<edgeBlocks, 256, 0, stream>>>(srcIdx, dstIdx, big0, agg, E, C1o);
    k_epilogue<<<N, C1o, 0, stream>>>(agg, big0, dinv, b1, big0, C1o);  // act1 = big0
  }

  // ---- layer 2: act1(256) @ W2 -> 128 ----
  {
    dim3 grid(rowBlocks, C2o / 16);
    k_gemm_wmma<256, 128><<<grid, 256, 0, stream>>>(big0, W2, dinv, big1, nRT);
    hipMemsetAsync(agg, 0, (size_t)N * C2o * sizeof(float), stream);
    k_scatter<4><<<edgeBlocks, 256, 0, stream>>>(srcIdx, dstIdx, big1, agg, E, C2o);
    k_epilogue<<<N, C2o, 0, stream>>>(agg, big1, dinv, b2, big1, C2o);  // act2 = big1
  }

  // ---- layer 3: act2(128) @ W3 -> 64, fused log_softmax ----
  {
    dim3 grid(rowBlocks, C3o / 16);
    k_gemm_wmma<128, 64><<<grid, 256, 0, stream>>>(big1, W3, dinv, big0, nRT);
    hipMemsetAsync(agg, 0, (size_t)N * C3o * sizeof(float), stream);
    k_scatter<2><<<edgeBlocks, 256, 0, stream>>>(srcIdx, dstIdx, big0, agg, E, C3o);
    k_epilogue_lsm<<<(N + WPB - 1) / WPB, 256, 0, stream>>>(agg, big0, dinv, b3, (float*)d_out, N);
  }
}